// MinJerkReg_3788161155713
// MI455X (gfx1250) — compile-verified
//
#include <hip/hip_runtime.h>

typedef float v2f __attribute__((ext_vector_type(2)));
typedef float v8f __attribute__((ext_vector_type(8)));

#define R1_BLOCKS  1024
#define R1_THREADS 256
#define MAX_SEG    128
#define NCH        4

// ---------------------------------------------------------------------------
// Kernel 1: regularizer trajectory reduction (bandwidth-bound: streams w_reg)
//   partial[b] = sum over steps n handled by block b of  w_reg[14+14n : 14+14n+14] . ref[:,n]
// ---------------------------------------------------------------------------
__global__ __launch_bounds__(R1_THREADS)
void reg_traj_kernel(const float* __restrict__ coeff,   // (4, num_seg, 8) flat
                     const float* __restrict__ ts,      // (num_seg+1,)
                     const float* __restrict__ w_reg,   // (14 + 14*num_steps,)
                     float* __restrict__ partials,      // (R1_BLOCKS,)
                     int num_steps, int num_seg)
{
    __shared__ float s_ts[MAX_SEG + 8];
    __shared__ float s_coef[NCH * MAX_SEG * 8];
    __shared__ float s_red[R1_THREADS];

    const int tid = threadIdx.x;
    for (int i = tid; i <= num_seg; i += R1_THREADS) s_ts[i] = ts[i];
    const int ncoef = NCH * num_seg * 8;
    for (int i = tid; i < ncoef; i += R1_THREADS) s_coef[i] = coeff[i];
    __syncthreads();

    const float ts0    = s_ts[0];
    const float tsN    = s_ts[num_seg];
    const float step   = (tsN - ts0) / (float)(num_steps - 1);
    const float invseg = (float)num_seg / (tsN - ts0);
    const int   cstride = num_seg * 8;

    float acc = 0.0f;
    const int gstride = gridDim.x * R1_THREADS;
    for (int n = blockIdx.x * R1_THREADS + tid; n < num_steps; n += gstride) {
        const float t = ts0 + step * (float)n;

        // searchsorted(ts[1:-1], t, 'left'): count of interior knots strictly < t
        int k = (int)((t - ts0) * invseg);
        if (k > num_seg - 1) k = num_seg - 1;
        if (k < 0) k = 0;
        while (k < num_seg - 1 && s_ts[k + 1] < t) ++k;
        while (k > 0 && s_ts[k] >= t) --k;
        const float dt = t - s_ts[k];

        float pw[8];
        pw[0] = 1.0f;
#pragma unroll
        for (int j = 1; j < 8; ++j) pw[j] = pw[j - 1] * dt;

        // basis rows: B_d[j] = falling(j,d) * dt^max(j-d,0)
        float B0[8], B1[8], B2[8], B3[8];
        const float c2t[8] = {0.f,0.f,2.f,6.f,12.f,20.f,30.f,42.f};
        const float c3t[8] = {0.f,0.f,0.f,6.f,24.f,60.f,120.f,210.f};
#pragma unroll
        for (int j = 0; j < 8; ++j) {
            B0[j] = pw[j];
            B1[j] = (float)j * pw[(j >= 1) ? j - 1 : 0];
            B2[j] = c2t[j]  * pw[(j >= 2) ? j - 2 : 0];
            B3[j] = c3t[j]  * pw[(j >= 3) ? j - 3 : 0];
        }

        float D0[4], D1[4], D2[4], D3[4];
#pragma unroll
        for (int a = 0; a < 4; ++a) {
            const float* cc = s_coef + a * cstride + k * 8;
            float d0 = 0.f, d1 = 0.f, d2 = 0.f, d3 = 0.f;
#pragma unroll
            for (int j = 0; j < 8; ++j) {
                const float cj = cc[j];
                d0 = fmaf(B0[j], cj, d0);
                d1 = fmaf(B1[j], cj, d1);
                d2 = fmaf(B2[j], cj, d2);
                d3 = fmaf(B3[j], cj, d3);
            }
            D0[a] = d0; D1[a] = d1; D2[a] = d2; D3[a] = d3;
        }

        // ref row order: pos(x,y,z) vel acc jerk yaw yaw_rate
        const float* wr = w_reg + 14 + (size_t)14 * (size_t)n;
        float c = 0.f;
        c = fmaf(wr[0],  D0[0], c); c = fmaf(wr[1],  D0[1], c); c = fmaf(wr[2],  D0[2], c);
        c = fmaf(wr[3],  D1[0], c); c = fmaf(wr[4],  D1[1], c); c = fmaf(wr[5],  D1[2], c);
        c = fmaf(wr[6],  D2[0], c); c = fmaf(wr[7],  D2[1], c); c = fmaf(wr[8],  D2[2], c);
        c = fmaf(wr[9],  D3[0], c); c = fmaf(wr[10], D3[1], c); c = fmaf(wr[11], D3[2], c);
        c = fmaf(wr[12], D0[3], c); c = fmaf(wr[13], D1[3], c);
        acc += c;
    }

    // deterministic block tree reduction
    s_red[tid] = acc;
    __syncthreads();
    for (int s = R1_THREADS / 2; s > 0; s >>= 1) {
        if (tid < s) s_red[tid] += s_red[tid + s];
        __syncthreads();
    }
    if (tid == 0) partials[blockIdx.x] = s_red[0];
}

// ---------------------------------------------------------------------------
// Kernel 2: quad = sum_p (C Q C^T)_pp via V_WMMA_F32_16X16X4_F32.
// C padded to 16 rows (rows 4..15 zeroed by a 0/1 VALU mask -- EXEC stays
// all-1s through the whole loop). One wave per 16-column tile of Q.
// DIM is a compile-time constant so every B element folds into the 24-bit
// immediate offset of global_load_b32 off a single pointer bumped once per
// unrolled step. Unroll x2 with dual accumulators (independent WMMA chains).
//
// A 16x4 layout: lane L holds row M=L%16; VGPR0 = K+2*half, VGPR1 = K+1+2*half.
// B 4x16:  VGPR r, lanes 0-15 -> row K=r,   lanes 16-31 -> K=r+2, N=lane%16.
// D 16x16: VGPR r, lanes 0-15 -> M=r, N=lane, lanes 16-31 -> M=r+8.
// ---------------------------------------------------------------------------
template <int DIM>
__global__ __launch_bounds__(32)
void quad_wmma_kernel(const float* __restrict__ C,   // (4, DIM) flat
                      const float* __restrict__ Q,   // (DIM, DIM) flat
                      float* __restrict__ partials)  // (DIM/16,)
{
    const int lane  = threadIdx.x;
    const int half  = lane >> 4;           // 0: lanes 0-15, 1: lanes 16-31
    const int mr    = lane & 15;           // A row / B,D column-in-tile
    const int n     = blockIdx.x * 16 + mr;
    const float amask = (mr < NCH) ? 1.0f : 0.0f;   // zero-pad rows 4..15
    const int rm    = (mr < NCH) ? mr : 0;          // always-valid row index

    const float* __restrict__ Arow = C + rm * DIM + 2 * half;
    const float* __restrict__ Bcol = Q + (2 * half) * DIM + n;

    v8f acc0 = {0.f,0.f,0.f,0.f,0.f,0.f,0.f,0.f};
    v8f acc1 = {0.f,0.f,0.f,0.f,0.f,0.f,0.f,0.f};

    constexpr int ITERS = DIM / 4;         // WMMA K-steps
#pragma unroll 4
    for (int it = 0; it < ITERS; it += 2) {
        v2f a0 = *(const v2f*)(Arow);      // contiguous pair, 8B aligned
        v2f a1 = *(const v2f*)(Arow + 4);
        v2f b0, b1;
        b0.x = Bcol[0];                    // imm offset 0
        b0.y = Bcol[DIM];                  // imm offset 4*DIM bytes
        b1.x = Bcol[4 * DIM];              // imm offset 16*DIM bytes
        b1.y = Bcol[5 * DIM];              // imm offset 20*DIM bytes
        a0 *= amask;
        a1 *= amask;
        acc0 = __builtin_amdgcn_wmma_f32_16x16x4_f32(
                   false, a0, false, b0, (short)0, acc0, false, false);
        acc1 = __builtin_amdgcn_wmma_f32_16x16x4_f32(
                   false, a1, false, b1, (short)0, acc1, false, false);
        Arow += 8;
        Bcol += 8 * DIM;
    }
    const v8f acc = acc0 + acc1;

    // rows 0..3 of D live in acc[0..3] of lanes 0..15
    float part = 0.0f;
    if (half == 0) {
        part = acc[0] * C[n]
             + acc[1] * C[DIM     + n]
             + acc[2] * C[2 * DIM + n]
             + acc[3] * C[3 * DIM + n];
    }
    // wave32 shuffle tree reduction (deterministic)
#pragma unroll
    for (int off = 16; off > 0; off >>= 1)
        part += __shfl_down(part, off, 32);
    if (lane == 0) partials[blockIdx.x] = part;
}

// Generic-dim fallback (same algorithm, runtime dim) -- only used if the
// problem shape ever differs from the reference's 4 x 1024 / 1024 x 1024.
__global__ __launch_bounds__(32)
void quad_wmma_kernel_gen(const float* __restrict__ C,
                          const float* __restrict__ Q,
                          float* __restrict__ partials, int dim)
{
    const int lane  = threadIdx.x;
    const int half  = lane >> 4;
    const int mr    = lane & 15;
    const int n     = blockIdx.x * 16 + mr;
    const float amask = (mr < NCH) ? 1.0f : 0.0f;
    const int rm    = (mr < NCH) ? mr : 0;

    const float* __restrict__ Arow = C + (size_t)rm * dim + 2 * half;
    const float* __restrict__ Bcol = Q + (size_t)(2 * half) * dim + n;

    v8f acc = {0.f,0.f,0.f,0.f,0.f,0.f,0.f,0.f};
    for (int k0 = 0; k0 < dim; k0 += 4) {
        v2f a = *(const v2f*)(Arow);
        v2f b;
        b.x = Bcol[0];
        b.y = Bcol[dim];
        a *= amask;
        acc = __builtin_amdgcn_wmma_f32_16x16x4_f32(
                  false, a, false, b, (short)0, acc, false, false);
        Arow += 4;
        Bcol += 4 * (size_t)dim;
    }

    float part = 0.0f;
    if (half == 0) {
        part = acc[0] * C[n]
             + acc[1] * C[dim     + n]
             + acc[2] * C[2 * dim + n]
             + acc[3] * C[3 * dim + n];
    }
#pragma unroll
    for (int off = 16; off > 0; off >>= 1)
        part += __shfl_down(part, off, 32);
    if (lane == 0) partials[blockIdx.x] = part;
}

// ---------------------------------------------------------------------------
// Kernel 3: deterministic finalize: out = quad + rho * (w.x0 + reg)
// ---------------------------------------------------------------------------
__global__ void finalize_kernel(const float* __restrict__ reg_partials, int nreg,
                                const float* __restrict__ quad_partials, int nquad,
                                const float* __restrict__ w_reg,
                                const float* __restrict__ x0, int nx0,
                                const float* __restrict__ rho,
                                float* __restrict__ out)
{
    if (threadIdx.x == 0 && blockIdx.x == 0) {
        float reg = 0.f;
        for (int i = 0; i < nreg; ++i) reg += reg_partials[i];
        float quad = 0.f;
        for (int i = 0; i < nquad; ++i) quad += quad_partials[i];
        float dx = 0.f;
        for (int i = 0; i < nx0; ++i) dx = fmaf(w_reg[i], x0[i], dx);
        out[0] = quad + rho[0] * (dx + reg);
    }
}

extern "C" void kernel_launch(void* const* d_in, const int* in_sizes, int n_in,
                              void* d_out, int out_size, void* d_ws, size_t ws_size,
                              hipStream_t stream) {
    const float* coeff = (const float*)d_in[0];  // (4,128,8) f32
    const float* cost  = (const float*)d_in[1];  // (1024,1024) f32
    const float* ts    = (const float*)d_in[2];  // (129,) f32
    const float* x0    = (const float*)d_in[3];  // (14,) f32
    const float* w_reg = (const float*)d_in[4];  // (14 + 14*N,) f32
    const float* rho   = (const float*)d_in[5];  // scalar f32

    int num_seg = in_sizes[2] - 1;               // 128
    if (num_seg > MAX_SEG) num_seg = MAX_SEG;
    const int dim       = in_sizes[0] / NCH;     // 1024 = num_seg*(order+1)
    const int num_steps = (in_sizes[4] - 14) / 14; // 1,000,000
    const int nx0       = in_sizes[3];           // 14
    const int ntiles    = dim / 16;              // 64

    float* ws    = (float*)d_ws;
    float* regp  = ws;                 // R1_BLOCKS partials
    float* quadp = ws + R1_BLOCKS;     // ntiles partials

    reg_traj_kernel<<<R1_BLOCKS, R1_THREADS, 0, stream>>>(
        coeff, ts, w_reg, regp, num_steps, num_seg);

    if (dim == 1024) {
        quad_wmma_kernel<1024><<<ntiles, 32, 0, stream>>>(coeff, cost, quadp);
    } else {
        quad_wmma_kernel_gen<<<ntiles, 32, 0, stream>>>(coeff, cost, quadp, dim);
    }

    finalize_kernel<<<1, 1, 0, stream>>>(regp, R1_BLOCKS, quadp, ntiles,
                                         w_reg, x0, nx0, rho, (float*)d_out);
}